// EM_rec_loss_23819888624148
// MI455X (gfx1250) — compile-verified
//
#include <hip/hip_runtime.h>

typedef __attribute__((ext_vector_type(2))) float v2f;
typedef __attribute__((ext_vector_type(8))) float v8f;

// Problem constants (b=8,f=4 -> N=32; ns=nb=7; 128x128 images)
constexpr int NB = 7, NS = 7, HW = 128 * 128, NN = 32;
constexpr int TP = 256;                            // pixels per LDS tile
constexpr int TILES_PER_WG = 4;                    // 1024 pixels per workgroup
constexpr int WG_PER_N = HW / (TP * TILES_PER_WG); // 16
constexpr int STR = 260;                           // LDS row stride; 260 % 64 == 4 -> conflict-free b64 fetch
constexpr int ARR = 8 * STR;                       // 7 data rows + 1 shared zero row per array

__global__ void em_zero_out(float* out) {
    if (threadIdx.x == 0 && blockIdx.x == 0) out[0] = 0.0f;
}

__global__ __launch_bounds__(256)
void em_rec_loss_kernel(const float* __restrict__ seg,   // [N,NB,HW]
                        const float* __restrict__ masks, // [N,NS,HW]
                        const float* __restrict__ rec,   // [N,NB,3,HW]
                        const float* __restrict__ tgt,   // [N,3,HW]
                        const float* __restrict__ mvis,  // [N,NS,HW]
                        const float* __restrict__ ai,    // [N,NS,NB]
                        float* __restrict__ out)
{
    __shared__ float lds[5 * ARR];
    float* Lm  = lds;             // mask bit      (rows = s)
    float* Lmv = lds + ARR;       // vis bit       (rows = s)
    float* LA  = lds + 2 * ARR;   // -log(seg)     (rows = b)
    float* LB  = lds + 3 * ARR;   // -log(1-seg)   (rows = b)
    float* LE  = lds + 4 * ARR;   // sum_c diff^2  (rows = b)

    const int tid        = threadIdx.x;
    const int n          = blockIdx.x / WG_PER_N;
    const int chunk_base = (blockIdx.x % WG_PER_N) * (TP * TILES_PER_WG);

    const int  lane = tid & 31;
    const int  wv   = tid >> 5;        // wave 0..7
    const bool hi   = lane >= 16;
    const int  rc   = lane & 15;       // row (s) for A operands == col (b) for B operands
    // Pad lanes (rc >= 7) read the shared zero row -> zero A-rows / zero B-cols, no multiplies.
    const int  rowoff = ((rc < NS) ? rc : 7) * STR;

    // Zero row 7 of each array once (never touched by staging; covered by first barrier).
    for (int i = tid; i < 5 * STR; i += 256) {
        const int a = i / STR;
        lds[a * ARR + 7 * STR + (i - a * STR)] = 0.0f;
    }

    v8f c1 = {};   //  m      x (-logp)
    v8f c2 = {};   // (1-m)   x (-log1mp)   (pad rows finite garbage, ai-weight = 0)
    v8f c3 = {};   //  mv     x E

    for (int t = 0; t < TILES_PER_WG; ++t) {
        const int pb = chunk_base + t * TP;

        // ---- stage one 256-pixel tile into LDS (coalesced, streaming loads) ----
        for (int i = tid; i < 7 * TP; i += 256) {
            const int r  = i >> 8;          // 0..6 (TP == 256)
            const int p  = i & (TP - 1);
            const int gp = pb + p;

            const float sg = __builtin_nontemporal_load(&seg[(n * NB + r) * HW + gp]);
            LA[r * STR + p] = fminf(-__logf(sg), 100.0f);          // -clip(log p, -100)
            LB[r * STR + p] = fminf(-__logf(1.0f - sg), 100.0f);   // -clip(log(1-p), -100)

            const float mk = __builtin_nontemporal_load(&masks[(n * NS + r) * HW + gp]);
            Lm[r * STR + p] = (mk > 0.5f) ? 1.0f : 0.0f;
            const float mv = __builtin_nontemporal_load(&mvis[(n * NS + r) * HW + gp]);
            Lmv[r * STR + p] = (mv > 0.5f) ? 1.0f : 0.0f;

            float e = 0.0f;
            #pragma unroll
            for (int c = 0; c < 3; ++c) {
                const float d = __builtin_nontemporal_load(&rec[((n * NB + r) * 3 + c) * HW + gp])
                              - tgt[(n * 3 + c) * HW + gp];   // tgt reused 7x -> keep cacheable
                e += d * d;
            }
            LE[r * STR + p] = e;
        }
        __syncthreads();

        // ---- WMMA phase: this wave contracts K-pixels [wv*32, wv*32+32) ----
        #pragma unroll
        for (int stp = 0; stp < 8; ++stp) {
            // f32 WMMA 16x16x4 operand layout: lanes 0-15 carry K={0,1}, lanes 16-31 carry K={2,3}
            const int kp  = wv * 32 + stp * 4 + (hi ? 2 : 0);
            const int off = rowoff + kp;                 // even -> 8B-aligned b64 LDS fetch
            const v2f am  = *(const v2f*)(Lm  + off);    // A: m[s, k..k+1]
            const v2f amv = *(const v2f*)(Lmv + off);    // A: mv[s, k..k+1]
            const v2f bA  = *(const v2f*)(LA  + off);    // B: -logp[k..k+1, b]
            const v2f bB  = *(const v2f*)(LB  + off);    // B: -log1mp[k..k+1, b]
            const v2f bE  = *(const v2f*)(LE  + off);    // B: E[k..k+1, b]
            v2f a1m; a1m.x = 1.0f - am.x; a1m.y = 1.0f - am.y;

            c1 = __builtin_amdgcn_wmma_f32_16x16x4_f32(false, am,  false, bA, (short)0, c1, false, false);
            c2 = __builtin_amdgcn_wmma_f32_16x16x4_f32(false, a1m, false, bB, (short)0, c2, false, false);
            c3 = __builtin_amdgcn_wmma_f32_16x16x4_f32(false, amv, false, bE, (short)0, c3, false, false);
        }
        __syncthreads();
    }

    // ---- weight by attn_index and reduce ----
    // C/D layout: VGPR r, lanes 0-15 -> (M=r, N=lane); lanes 16-31 -> (M=r+8, N=lane-16)
    float part = 0.0f;
    #pragma unroll
    for (int r = 0; r < 8; ++r) {
        const int s = r + (hi ? 8 : 0);
        float aiv = 0.0f;
        if (s < NS && rc < NB) aiv = ai[(n * NS + s) * NB + rc];
        part += aiv * ((c1[r] + c2[r]) * (1.0f / HW) + 0.1f * c3[r]);
    }
    #pragma unroll
    for (int m = 16; m > 0; m >>= 1) part += __shfl_xor(part, m, 32);
    if (lane == 0) atomicAdd(out, part * (20.0f / (NN * NS * NB)));
}

extern "C" void kernel_launch(void* const* d_in, const int* in_sizes, int n_in,
                              void* d_out, int out_size, void* d_ws, size_t ws_size,
                              hipStream_t stream) {
    const float* seg   = (const float*)d_in[0];
    const float* masks = (const float*)d_in[1];
    const float* rec   = (const float*)d_in[2];
    const float* tgt   = (const float*)d_in[3];
    const float* mvis  = (const float*)d_in[4];
    const float* ai    = (const float*)d_in[5];
    float* out = (float*)d_out;

    em_zero_out<<<1, 32, 0, stream>>>(out);
    em_rec_loss_kernel<<<NN * WG_PER_N, 256, 0, stream>>>(seg, masks, rec, tgt, mvis, ai, out);
    (void)in_sizes; (void)n_in; (void)out_size; (void)d_ws; (void)ws_size;
}